// MTxyt_Lookup_Class_57999238365242
// MI455X (gfx1250) — compile-verified
//
#include <hip/hip_runtime.h>
#include <hip/hip_bf16.h>

typedef __attribute__((ext_vector_type(2))) float v2f;
typedef __attribute__((ext_vector_type(8))) float v8f;

#define HAVE_WMMA_F32X4 __has_builtin(__builtin_amdgcn_wmma_f32_16x16x4_f32)

// Problem constants (from reference)
#define NMODELS 64
#define NREC    32
#define NTINT   614            // int(512*24/20)
#define KT      (3*NTINT)      // 1842 flattened (k,t)
#define NTILE   116            // ceil(1842/16)
#define ROWP    (NTILE*16)     // 1856 padded row stride per (m,r) pair
#define CELL    (3*NTINT*6)    // 11052 floats per (i,j) table cell

// ---------------------------------------------------------------------------
// Kernel 1: bilinear 4-corner gather + einsum over (corner, 6 coeffs)
// One wave (32 lanes) per (m,r) pair. Uses V_WMMA_F32_16X16X4_F32:
//   A: 16 kt-rows x 4 coeffs  (lanes 0-15: K=0,1 ; lanes 16-31: K=2,3)
//   B: weight-scaled mt replicated across all 16 columns -> every D column
//      equals the desired dot; accumulate C across 4 corners x 2 K-passes.
// ---------------------------------------------------------------------------
__global__ __launch_bounds__(32)
void k_einsum(const float* __restrict__ model, const float* __restrict__ recv,
              const float* __restrict__ table, float* __restrict__ y)
{
    const int pair = blockIdx.x;          // 0..2047
    const int m = pair >> 5;              // /NREC
    const int r = pair & 31;
    const int lane = threadIdx.x;
    const bool hi = lane >= 16;

    const float* mp = model + m * 9;
    const float hx = mp[0] - recv[r * 2 + 0];
    const float hy = mp[1] - recv[r * 2 + 1];
    const float inv_d = 63.0f * 0.5f;     // (NX-1)/(GRID_MAX-GRID_MIN)
    const float u = (hx + 1.0f) * inv_d;
    const float v = (hy + 1.0f) * inv_d;
    int i0 = (int)floorf(u); i0 = i0 < 0 ? 0 : (i0 > 62 ? 62 : i0);
    int j0 = (int)floorf(v); j0 = j0 < 0 ? 0 : (j0 > 62 ? 62 : j0);
    const float fu = u - (float)i0, fv = v - (float)j0;

    float w[4];
    w[0] = (1.0f - fu) * (1.0f - fv);
    w[1] = fu * (1.0f - fv);
    w[2] = (1.0f - fu) * fv;
    w[3] = fu * fv;

    const float* base[4];
    base[0] = table + (size_t)(i0 * 64 + j0) * CELL;
    base[1] = table + (size_t)((i0 + 1) * 64 + j0) * CELL;
    base[2] = table + (size_t)(i0 * 64 + (j0 + 1)) * CELL;
    base[3] = table + (size_t)((i0 + 1) * 64 + (j0 + 1)) * CELL;

    float mt[6];
    #pragma unroll
    for (int j = 0; j < 6; ++j) mt[j] = mp[3 + j];

    // B operands: column-replicated q = w_c * mt (pass1: q0..q3, pass2: q4,q5,0,0)
    v2f b1[4], b2[4];
    #pragma unroll
    for (int c = 0; c < 4; ++c) {
        const float q0 = w[c] * mt[0], q1 = w[c] * mt[1], q2 = w[c] * mt[2];
        const float q3 = w[c] * mt[3], q4 = w[c] * mt[4], q5 = w[c] * mt[5];
        b1[c].x = hi ? q2 : q0;  b1[c].y = hi ? q3 : q1;
        b2[c].x = hi ? 0.f : q4; b2[c].y = hi ? 0.f : q5;
    }

    const int rowl = lane & 15;
    const int koff = hi ? 2 : 0;
    float* yrow = y + (size_t)pair * ROWP;

    for (int tile = 0; tile < NTILE; ++tile) {
        int row = tile * 16 + rowl;
        if (row > KT - 1) row = KT - 1;          // clamp tail tile, EXEC stays all-1s
        v8f acc = {0.f, 0.f, 0.f, 0.f, 0.f, 0.f, 0.f, 0.f};
#if HAVE_WMMA_F32X4
        #pragma unroll
        for (int c = 0; c < 4; ++c) {
            const float* p = base[c] + row * 6;
            v2f a1 = *(const v2f*)(p + koff);    // K0,K1 (lo) / K2,K3 (hi)
            v2f a2r = *(const v2f*)(p + 4);      // coeffs 4,5
            v2f a2; a2.x = hi ? 0.f : a2r.x; a2.y = hi ? 0.f : a2r.y;
            acc = __builtin_amdgcn_wmma_f32_16x16x4_f32(
                false, a1, false, b1[c], (short)0, acc, false, false);
            acc = __builtin_amdgcn_wmma_f32_16x16x4_f32(
                false, a2, false, b2[c], (short)0, acc, false, false);
            if (tile + 1 < NTILE)
                __builtin_prefetch(base[c] + (tile * 16 + 16 + rowl) * 6, 0, 1);
        }
        // D columns are all equal; lane 0 holds rows kt..kt+7, lane 16 rows kt+8..kt+15
        if (lane == 0 || lane == 16) {
            const int o = tile * 16 + (hi ? 8 : 0);
            float4 lo4 = make_float4(acc[0], acc[1], acc[2], acc[3]);
            float4 hi4 = make_float4(acc[4], acc[5], acc[6], acc[7]);
            *(float4*)(yrow + o) = lo4;
            *(float4*)(yrow + o + 4) = hi4;
        }
#else
        // VALU fallback: lanes 0-15 each compute one row's 24-length dot
        float s = 0.f;
        #pragma unroll
        for (int c = 0; c < 4; ++c) {
            const float* p = base[c] + row * 6;
            #pragma unroll
            for (int j = 0; j < 6; ++j) s += (w[c] * mt[j]) * p[j];
        }
        if (!hi) yrow[tile * 16 + rowl] = s;
#endif
    }
}

// ---------------------------------------------------------------------------
// Kernel 2: not-a-knot spline second derivatives, one thread per row (6144).
// Row reduction: x1 = r1/6, x[n-2] = r[n-2]/6; interior i=2..611 is a
// constant-coefficient [1,4,1] tridiagonal -> Thomas with LDS-cached cp[].
// ---------------------------------------------------------------------------
__global__ __launch_bounds__(256)
void k_spline(const float* __restrict__ y, float* __restrict__ mm)
{
    __shared__ float cp[612];
    if (threadIdx.x == 0) {
        cp[2] = 0.25f;
        for (int i = 3; i <= 611; ++i) cp[i] = 1.0f / (4.0f - cp[i - 1]);
    }
    __syncthreads();

    const int rowid = blockIdx.x * blockDim.x + threadIdx.x;  // 0..6143
    const int pair = rowid / 3, k = rowid % 3;
    const float* yr = y + (size_t)pair * ROWP + k * NTINT;
    float* mr = mm + (size_t)pair * ROWP + k * NTINT;

    const float inv_h2 = (613.0f / 24.0f) * (613.0f / 24.0f); // 1/H_INT^2

    float ya = yr[0], yb = yr[1], yc = yr[2];
    const float x1   = (ya - 2.0f * yb + yc) * inv_h2;                 // r1/6
    const float x612 = (yr[611] - 2.0f * yr[612] + yr[613]) * inv_h2;  // r612/6

    // forward sweep
    ya = yb; yb = yc; yc = yr[3];
    float dp = (6.0f * (ya - 2.0f * yb + yc) * inv_h2 - x1) * 0.25f;   // i=2
    mr[2] = dp;
    for (int i = 3; i <= 610; ++i) {
        ya = yb; yb = yc; yc = yr[i + 1];
        const float ri = 6.0f * (ya - 2.0f * yb + yc) * inv_h2;
        dp = (ri - dp) * cp[i];
        mr[i] = dp;
    }
    ya = yb; yb = yc; yc = yr[612];                                    // i=611
    const float r611 = 6.0f * (ya - 2.0f * yb + yc) * inv_h2;
    dp = (r611 - x612 - dp) * cp[611];

    // back substitution
    float x = dp;
    const float x611 = x;
    mr[611] = x;
    for (int i = 610; i >= 2; --i) {
        const float dpi = mr[i];
        x = dpi - cp[i] * x;
        mr[i] = x;
    }
    // x now holds x2
    mr[1]   = x1;
    mr[612] = x612;
    mr[0]   = 2.0f * x1 - x;          // row0: x0 - 2x1 + x2 = 0
    mr[613] = 2.0f * x612 - x611;     // rowN: x611 - 2x612 + x613 = 0
}

// ---------------------------------------------------------------------------
// Kernel 3: shifted cubic-spline evaluation at 512 output times.
// grid = (2, 96, 64) x 256 threads ; out shape (M, R, 3, 512)
// ---------------------------------------------------------------------------
__global__ __launch_bounds__(256)
void k_eval(const float* __restrict__ model, const float* __restrict__ y,
            const float* __restrict__ mm, float* __restrict__ out)
{
    const int n = blockIdx.x * 256 + threadIdx.x;  // 0..511
    const int rk = blockIdx.y;                     // 0..95
    const int m = blockIdx.z;                      // 0..63
    const int r = rk / 3, k = rk % 3;

    const float shift = model[m * 9 + 2];
    const float t_out = (20.0f / 511.0f) * (float)n;
    const float H = 24.0f / 613.0f;                // H_INT
    const float u = (t_out - shift + 2.0f) / H;    // (tq - T0_INT)/H
    int idx = (int)floorf(u);
    idx = idx < 0 ? 0 : (idx > NTINT - 2 ? NTINT - 2 : idx);
    const float a = (float)(idx + 1) - u;
    const float b = u - (float)idx;

    const size_t base = (size_t)(m * 32 + r) * ROWP + k * NTINT;
    const float yi  = y[base + idx],  yi1 = y[base + idx + 1];
    const float mi  = mm[base + idx], mi1 = mm[base + idx + 1];
    const float h26 = H * H / 6.0f;

    const float val = a * yi + b * yi1 +
                      ((a * a * a - a) * mi + (b * b * b - b) * mi1) * h26;
    out[(((size_t)m * 32 + r) * 3 + k) * 512 + n] = val;
}

// ---------------------------------------------------------------------------
extern "C" void kernel_launch(void* const* d_in, const int* in_sizes, int n_in,
                              void* d_out, int out_size, void* d_ws, size_t ws_size,
                              hipStream_t stream)
{
    const float* model = (const float*)d_in[0];   // (64, 9)
    const float* recv  = (const float*)d_in[1];   // (32, 2)
    const float* table = (const float*)d_in[2];   // (64, 64, 3, 614, 6)
    float* out = (float*)d_out;                   // (64, 32, 3, 512)

    float* wsf = (float*)d_ws;
    float* y  = wsf;                              // (2048, 1856)
    float* mm = wsf + (size_t)2048 * ROWP;        // (2048, 1856)

    k_einsum<<<2048, 32, 0, stream>>>(model, recv, table, y);
    k_spline<<<24, 256, 0, stream>>>(y, mm);
    k_eval<<<dim3(2, 96, 64), 256, 0, stream>>>(model, y, mm, out);
}